// Model_39676907884503
// MI455X (gfx1250) — compile-verified
//
#include <hip/hip_runtime.h>

// Problem constants (B,H,S,D) = (8,16,256,256), S == D
#define BQ 8
#define HQ 16
#define SQ 256
#define DQ 256

typedef __attribute__((ext_vector_type(2))) float v2f;
typedef __attribute__((ext_vector_type(8))) float v8f;
typedef __attribute__((ext_vector_type(4))) int   v4i;

#define LQK_STRIDE 258   // 64-row stripe buffer, +2 pad => conflict-free A-frag reads
#define KSTG_STRIDE 20   // 256 rows x 16 d, +4 pad => 16B-aligned b128s, conflict-free B reads

// ---- CDNA5 async global->LDS staging (ASYNCcnt) with safe fallback ----
#if defined(__AMDGCN__) && __has_builtin(__builtin_amdgcn_global_load_async_to_lds_b128)
#define USE_ASYNC_LDS 1
#else
#define USE_ASYNC_LDS 0
#endif

#if USE_ASYNC_LDS && __has_builtin(__builtin_amdgcn_s_wait_asynccnt)
#define WAIT_ASYNC(n) __builtin_amdgcn_s_wait_asynccnt(n)
#elif USE_ASYNC_LDS
#define WAIT_ASYNC(n) asm volatile("s_wait_asynccnt %0" ::"n"(n) : "memory")
#else
#define WAIT_ASYNC(n) do {} while (0)
#endif

// Builtin signature (from clang diagnostic): param0 is v4i in addrspace(1)
// (printed as '__device__ *' in HIP), so pass typed int4 pointers:
typedef __attribute__((address_space(1))) v4i* gpv4_t;   // global src
typedef __attribute__((address_space(3))) v4i* lpv4_t;   // LDS dst (__shared__)

// Move 16 bytes global -> LDS. Async path: one GLOBAL_LOAD_ASYNC_TO_LDS_B128
// per lane (no VGPR data round-trip, tracked by ASYNCcnt). Generic->AS casts
// go through integers: LDS generic pointers keep the LDS byte offset in
// bits [31:0] (ISA aperture rules), so the truncating cast is exact.
static __device__ __forceinline__ void cp16(const float* g, float* l) {
#if USE_ASYNC_LDS
  __builtin_amdgcn_global_load_async_to_lds_b128(
      (gpv4_t)(uintptr_t)g, (lpv4_t)(uint32_t)(uintptr_t)l, 0, 0);
#else
  *(float4*)l = *(const float4*)g;
#endif
}

static __device__ __forceinline__ v8f wmma_f32(v2f a, v2f b, v8f c) {
  return __builtin_amdgcn_wmma_f32_16x16x4_f32(false, a, false, b, (short)0, c, false, false);
}

// V staging: chunk c = t-rows [16c,16c+16) x all 256 dd -> [16][256] tile
static __device__ __forceinline__ void issueV(const float* __restrict__ vb,
                                              float* __restrict__ buf,
                                              int tid, int c) {
#pragma unroll
  for (int j = 0; j < 8; ++j) {
    int f = tid + 128 * j;
    int i = f >> 6, col = (f & 63) * 4;
    cp16(vb + (c * 16 + i) * DQ + col, buf + i * 256 + col);
  }
}

// K staging: chunk c = all 256 t-rows x d in [16c,16c+16) -> [256][16+pad] tile
static __device__ __forceinline__ void issueK(const float* __restrict__ kb,
                                              float* __restrict__ buf,
                                              int tid, int c) {
#pragma unroll
  for (int j = 0; j < 8; ++j) {
    int f = tid + 128 * j;
    int trow = f >> 2, j4 = f & 3;
    cp16(kb + trow * DQ + c * 16 + j4 * 4, buf + trow * KSTG_STRIDE + j4 * 4);
  }
}

// attn-stripe x V matmul: acc[nn] += lqk_stripe(16x256) * V(256x256), reduction over t.
// Shared by phase 2 (qk*V) and phase 4 (p*V). Double-buffered async staging.
static __device__ __forceinline__ void mm_stripe_x_V(
    const float* __restrict__ vb, float* __restrict__ s0, float* __restrict__ s1,
    const float* __restrict__ lqk, int tid, int l16, int half, int lrow0,
    v8f acc[16])
{
  issueV(vb, s0, tid, 0);
  for (int c = 0; c < 16; ++c) {
    float* cur = (c & 1) ? s1 : s0;
    float* nxt = (c & 1) ? s0 : s1;
    if (c + 1 < 16) {           // fill next buffer while chunk c computes
      issueV(vb, nxt, tid, c + 1);
      WAIT_ASYNC(8);            // own chunk-c async ops retired (in-order)
    } else {
      WAIT_ASYNC(0);
    }
    __syncthreads();            // all waves' chunk-c data visible in LDS
#pragma unroll
    for (int sub = 0; sub < 4; ++sub) {    // 4 k-steps of 4 per chunk
      const int kloc = sub * 4 + half * 2;
      // A frag: stripe rows (wave-private): a = lqk[m][t0..t0+1]
      v2f a = *(const v2f*)(lqk + (lrow0 + l16) * LQK_STRIDE + c * 16 + kloc);
#pragma unroll
      for (int nn = 0; nn < 16; ++nn) {    // 16 N-tiles over dd
        v2f b;
        b.x = cur[kloc * 256 + nn * 16 + l16];
        b.y = cur[(kloc + 1) * 256 + nn * 16 + l16];
        acc[nn] = wmma_f32(a, b, acc[nn]);
      }
    }
    __syncthreads();            // chunk-c buffer reusable next iteration
  }
}

__global__ __launch_bounds__(128, 1) void fused_mask_attn(
    const float* __restrict__ q, const float* __restrict__ k,
    const float* __restrict__ v, const float* __restrict__ mask,
    float* __restrict__ out)
{
  __shared__ float lqk[64 * LQK_STRIDE];        // qk / p stripes (64-row block)
  __shared__ float stg0[SQ * KSTG_STRIDE];      // staging double buffer:
  __shared__ float stg1[SQ * KSTG_STRIDE];      // K [256][20] or V [16][256]

  const int tid  = threadIdx.x;
  const int lane = tid & 31;
  const int l16  = lane & 15;
  const int half = lane >> 4;
  const int wave = tid >> 5;

  const int rb = blockIdx.x & 3;                // 64-row block within S
  const int bh = blockIdx.x >> 2;               // fused (b,h)
  const int h  = bh & (HQ - 1);

  const float* qb = q    + (size_t)bh * SQ * DQ;
  const float* kb = k    + (size_t)bh * SQ * DQ;
  const float* vb = v    + (size_t)bh * SQ * DQ;
  const float* mb = mask + (size_t)h * SQ * SQ;
  float*       ob = out  + (size_t)bh * SQ * DQ;

  const int lrow0 = wave * 16;                  // stripe base (local, 0..48)
  const int srow0 = rb * 64 + lrow0;            // stripe base (global s)

  // ---------------- Phase 1: qk = Q * K^T (reduction over d) ----------------
  v8f acc[16];
#pragma unroll
  for (int t = 0; t < 16; ++t) acc[t] = {};

  issueK(kb, stg0, tid, 0);
  for (int c = 0; c < 16; ++c) {
    float* cur = (c & 1) ? stg1 : stg0;
    float* nxt = (c & 1) ? stg0 : stg1;
    if (c + 1 < 16) {
      issueK(kb, nxt, tid, c + 1);
      WAIT_ASYNC(8);
    } else {
      WAIT_ASYNC(0);
    }
    __syncthreads();
#pragma unroll
    for (int sub = 0; sub < 4; ++sub) {
      const int dloc = sub * 4 + half * 2;
      // A frag from Q (global, reused across all 16 N-tiles)
      v2f a = *(const v2f*)(qb + (srow0 + l16) * DQ + c * 16 + dloc);
#pragma unroll
      for (int tt = 0; tt < 16; ++tt) {         // B frag: B[d][t] = k[t][d]
        v2f b = *(const v2f*)(cur + (tt * 16 + l16) * KSTG_STRIDE + dloc);
        acc[tt] = wmma_f32(a, b, acc[tt]);
      }
    }
    __syncthreads();
  }

  // scale + additive mask, spill stripe to LDS in A-fragment-friendly layout
  const float scale = 0.0625f;                  // 1/sqrt(256)
#pragma unroll
  for (int tt = 0; tt < 16; ++tt) {
#pragma unroll
    for (int r = 0; r < 8; ++r) {
      const int lr  = lrow0 + r + half * 8;
      const int col = tt * 16 + l16;
      float m = mb[(rb * 64 + lr) * SQ + col];
      lqk[lr * LQK_STRIDE + col] = acc[tt][r] * scale + m;
    }
  }

  // ---------------- Phase 2: a1 = qk * V (reduction over t) ----------------
  v8f acc2[16];
#pragma unroll
  for (int t = 0; t < 16; ++t) acc2[t] = {};
  mm_stripe_x_V(vb, stg0, stg1, lqk, tid, l16, half, lrow0, acc2);

  // ---------------- Phase 3: mask-fill + row softmax ----------------
#pragma unroll
  for (int nn = 0; nn < 16; ++nn) {
#pragma unroll
    for (int r = 0; r < 8; ++r) {
      const int col  = nn * 16 + l16;
      const int srow = rb * 64 + lrow0 + r + half * 8;
      float m = mb[srow * SQ + col];
      acc2[nn][r] = (m == 0.0f) ? -1.0e9f : acc2[nn][r];
    }
  }
  float mx[8], inv[8];
#pragma unroll
  for (int r = 0; r < 8; ++r) {
    float x = acc2[0][r];
#pragma unroll
    for (int nn = 1; nn < 16; ++nn) x = fmaxf(x, acc2[nn][r]);
    x = fmaxf(x, __shfl_xor(x, 1, 32));
    x = fmaxf(x, __shfl_xor(x, 2, 32));
    x = fmaxf(x, __shfl_xor(x, 4, 32));
    x = fmaxf(x, __shfl_xor(x, 8, 32));         // reduce within 16-lane half (wave32)
    mx[r] = x;
  }
#pragma unroll
  for (int r = 0; r < 8; ++r) {
    float s = 0.0f;
#pragma unroll
    for (int nn = 0; nn < 16; ++nn) {
      float e = __expf(acc2[nn][r] - mx[r]);
      acc2[nn][r] = e;
      s += e;
    }
    s += __shfl_xor(s, 1, 32);
    s += __shfl_xor(s, 2, 32);
    s += __shfl_xor(s, 4, 32);
    s += __shfl_xor(s, 8, 32);
    inv[r] = 1.0f / s;
  }
#pragma unroll
  for (int nn = 0; nn < 16; ++nn) {             // normalized p -> LDS (wave-private rows)
#pragma unroll
    for (int r = 0; r < 8; ++r) {
      const int lr  = lrow0 + r + half * 8;
      const int col = nn * 16 + l16;
      lqk[lr * LQK_STRIDE + col] = acc2[nn][r] * inv[r];
    }
  }

  // ---------------- Phase 4: out = p * V ----------------
  v8f acc3[16];
#pragma unroll
  for (int t = 0; t < 16; ++t) acc3[t] = {};
  mm_stripe_x_V(vb, stg0, stg1, lqk, tid, l16, half, lrow0, acc3);

#pragma unroll
  for (int nn = 0; nn < 16; ++nn) {
#pragma unroll
    for (int r = 0; r < 8; ++r) {
      const int srow = rb * 64 + lrow0 + r + half * 8;
      const int col  = nn * 16 + l16;
      ob[srow * DQ + col] = acc3[nn][r];
    }
  }
}

extern "C" void kernel_launch(void* const* d_in, const int* in_sizes, int n_in,
                              void* d_out, int out_size, void* d_ws, size_t ws_size,
                              hipStream_t stream) {
  (void)in_sizes; (void)n_in; (void)out_size; (void)d_ws; (void)ws_size;
  const float* q    = (const float*)d_in[0];
  const float* k    = (const float*)d_in[1];
  const float* v    = (const float*)d_in[2];
  const float* mask = (const float*)d_in[3];
  float* out = (float*)d_out;
  dim3 grid(BQ * HQ * (SQ / 64));               // 512 workgroups
  dim3 block(128);                              // 4 waves (wave32)
  hipLaunchKernelGGL(fused_mask_attn, grid, block, 0, stream, q, k, v, mask, out);
}